// DynamicHierarchicalVQ_3917010174115
// MI455X (gfx1250) — compile-verified
//
#include <hip/hip_runtime.h>
#include <hip/hip_bf16.h>

typedef __attribute__((ext_vector_type(2))) float v2f;
typedef __attribute__((ext_vector_type(4))) float v4f;
typedef __attribute__((ext_vector_type(8))) float v8f;

// Problem shape
#define N_ROWS 32768        // B*S = 16*2048
#define DIM    512          // 2*L
#define K_SYM  1024
#define K_CON  256

// Output layout (floats), in reference return order
#define O_CPLX  ((size_t)0)                    // 16.78M (interleaved re,im)
#define O_SYMP  ((size_t)16777216)             // 33.55M one-hot
#define O_CONP  ((size_t)50331648)             // 8.39M one-hot
#define O_LSYM  ((size_t)58720256)
#define O_LCON  ((size_t)58720257)
#define O_SIDX  ((size_t)58720258)
#define O_CIDX  ((size_t)58753026)
#define O_CONF  ((size_t)58785794)

// ---------------- row / codebook norms ----------------
__global__ __launch_bounds__(256) void row_norms_k(const float* __restrict__ zr,
                                                   const float* __restrict__ zi,
                                                   float* __restrict__ zn2) {
  __shared__ float red[256];
  const int n = blockIdx.x, t = threadIdx.x;
  float a = zr[(size_t)n * 256 + t];
  float b = zi[(size_t)n * 256 + t];
  red[t] = a * a + b * b;
  __syncthreads();
  for (int s = 128; s > 0; s >>= 1) { if (t < s) red[t] += red[t + s]; __syncthreads(); }
  if (t == 0) zn2[n] = red[0];
}

__global__ __launch_bounds__(256) void cb_norms_k(const float* __restrict__ cb,
                                                  float* __restrict__ cn2) {
  __shared__ float red[256];
  const int n = blockIdx.x, t = threadIdx.x;
  float a = cb[(size_t)n * 512 + t];
  float b = cb[(size_t)n * 512 + 256 + t];
  red[t] = a * a + b * b;
  __syncthreads();
  for (int s = 128; s > 0; s >>= 1) { if (t < s) red[t] += red[t + s]; __syncthreads(); }
  if (t == 0) cn2[n] = red[0];
}

// ---------------- stage-1: fp32 WMMA GEMM + fused argmin ----------------
// Block: 256 threads = 8 waves; covers 32 rows x 1024 codes.
// Wave w handles codes [w*128, w*128+128): 2 row-tiles x 8 code-tiles of 16x16.
__global__ __launch_bounds__(256) void sym_gemm_k(const float* __restrict__ zr,
                                                  const float* __restrict__ zi,
                                                  const float* __restrict__ cb,
                                                  const float* __restrict__ cn2,
                                                  const float* __restrict__ zn2,
                                                  int*  __restrict__ sym_idx,
                                                  float* __restrict__ min_dist) {
  __shared__ float zt[32 * 516];       // 32 rows x 512, stride padded to 516 (bank-conflict free)
  __shared__ float s_val[8][32];
  __shared__ int   s_idx[8][32];

  const int tid  = threadIdx.x;
  const int wave = tid >> 5, lane = tid & 31;
  const int l16  = lane & 15, half = lane >> 4;
  const int row0 = blockIdx.x * 32;

  // stage z tile (concat handled here: d<256 -> real, else imag)
  for (int i = tid; i < 32 * 512; i += 256) {
    const int r = i >> 9, c = i & 511;
    const float v = (c < 256) ? zr[(size_t)(row0 + r) * 256 + c]
                              : zi[(size_t)(row0 + r) * 256 + (c - 256)];
    zt[r * 516 + c] = v;
  }
  __syncthreads();

  const int code0 = wave * 128;
  float cn2r[8];
#pragma unroll
  for (int t = 0; t < 8; ++t) cn2r[t] = cn2[code0 + t * 16 + l16];

  const v8f vzero = {0.f, 0.f, 0.f, 0.f, 0.f, 0.f, 0.f, 0.f};
  v8f acc[2][8];
#pragma unroll
  for (int rt = 0; rt < 2; ++rt)
#pragma unroll
    for (int t = 0; t < 8; ++t) acc[rt][t] = vzero;

  const int khalf = half * 2;   // A/B fragment: lane holds K = khalf, khalf+1
  for (int k = 0; k < 512; k += 4) {
    const int kk = k + khalf;
    const v2f a0 = *(const v2f*)&zt[(0 * 16 + l16) * 516 + kk];
    const v2f a1 = *(const v2f*)&zt[(1 * 16 + l16) * 516 + kk];
#pragma unroll
    for (int t = 0; t < 8; ++t) {
      const v2f b = *(const v2f*)(cb + (size_t)(code0 + t * 16 + l16) * 512 + kk);
      acc[0][t] = __builtin_amdgcn_wmma_f32_16x16x4_f32(false, a0, false, b,
                                                        (short)0, acc[0][t], false, false);
      acc[1][t] = __builtin_amdgcn_wmma_f32_16x16x4_f32(false, a1, false, b,
                                                        (short)0, acc[1][t], false, false);
    }
  }

  // per-lane argmin of (||c||^2 - 2 z.c) over this wave's 128 codes, per row
#pragma unroll
  for (int rt = 0; rt < 2; ++rt) {
#pragma unroll
    for (int r = 0; r < 8; ++r) {
      float best = 3.4e38f;
      int bestn = 0x7fffffff;
#pragma unroll
      for (int t = 0; t < 8; ++t) {
        const float q = cn2r[t] - 2.0f * acc[rt][t][r];
        const int n = code0 + t * 16 + l16;
        if (q < best || (q == best && n < bestn)) { best = q; bestn = n; }
      }
      // butterfly min across the 16 lanes holding this row (masks<16 stay in-half)
      for (int m = 1; m < 16; m <<= 1) {
        const float ov = __shfl_xor(best, m, 32);
        const int   oi = __shfl_xor(bestn, m, 32);
        if (ov < best || (ov == best && oi < bestn)) { best = ov; bestn = oi; }
      }
      if (l16 == 0) {
        const int lr = rt * 16 + r + 8 * half;   // local row 0..31
        s_val[wave][lr] = best;
        s_idx[wave][lr] = bestn;
      }
    }
  }
  __syncthreads();

  if (tid < 32) {
    float best = s_val[0][tid];
    int bestn = s_idx[0][tid];
#pragma unroll
    for (int w = 1; w < 8; ++w) {
      const float v = s_val[w][tid];
      const int i2 = s_idx[w][tid];
      if (v < best || (v == best && i2 < bestn)) { best = v; bestn = i2; }
    }
    const int row = row0 + tid;
    sym_idx[row] = bestn;
    min_dist[row] = zn2[row] + best;   // expanded-form distance (matches reference)
  }
}

// ---------------- stage-2: 1024x256 exact distance table + per-symbol argmin ----------------
__global__ __launch_bounds__(256) void concept_table_k(const float* __restrict__ scb,
                                                       const float* __restrict__ ccb,
                                                       int*  __restrict__ con_arg,
                                                       float* __restrict__ con_min) {
  __shared__ float sC[64 * 257];   // [dim][concept], padded
  __shared__ float sS[16 * 64];    // [symbol][dim]
  __shared__ float red_v[256];
  __shared__ int   red_i[256];
  const int t = threadIdx.x;       // concept id within block
  const int j0 = blockIdx.x * 16;  // 16 symbols per block

  float acc[16];
#pragma unroll
  for (int j = 0; j < 16; ++j) acc[j] = 0.f;

  for (int d0 = 0; d0 < 512; d0 += 64) {
    __syncthreads();
    for (int i = t; i < 64 * 256; i += 256) {
      const int k = i >> 6, dd = i & 63;
      sC[dd * 257 + k] = ccb[(size_t)k * 512 + d0 + dd];
    }
    for (int i = t; i < 16 * 64; i += 256) {
      const int j = i >> 6, dd = i & 63;
      sS[j * 64 + dd] = scb[(size_t)(j0 + j) * 512 + d0 + dd];
    }
    __syncthreads();
    for (int dd = 0; dd < 64; ++dd) {
      const float c = sC[dd * 257 + t];
#pragma unroll
      for (int j = 0; j < 16; ++j) {
        const float diff = sS[j * 64 + dd] - c;
        acc[j] += diff * diff;
      }
    }
  }

  for (int j = 0; j < 16; ++j) {
    __syncthreads();
    red_v[t] = acc[j];
    red_i[t] = t;
    __syncthreads();
    for (int s = 128; s > 0; s >>= 1) {
      if (t < s) {
        if (red_v[t + s] < red_v[t] ||
            (red_v[t + s] == red_v[t] && red_i[t + s] < red_i[t])) {
          red_v[t] = red_v[t + s];
          red_i[t] = red_i[t + s];
        }
      }
      __syncthreads();
    }
    if (t == 0) { con_arg[j0 + j] = red_i[0]; con_min[j0 + j] = red_v[0]; }
  }
}

// ---------------- outputs: gather/scatter + one-hots + loss partials ----------------
__global__ __launch_bounds__(256) void scatter_out_k(const float* __restrict__ zr,
                                                     const float* __restrict__ zi,
                                                     const float* __restrict__ scb,
                                                     const int*  __restrict__ sym_idx,
                                                     const float* __restrict__ min_dist,
                                                     const int*  __restrict__ con_arg,
                                                     const float* __restrict__ con_min,
                                                     float* __restrict__ out,
                                                     float* __restrict__ part_sym,
                                                     float* __restrict__ part_con) {
  const int n = blockIdx.x, t = threadIdx.x;
  __shared__ int sh_j, sh_ck;
  __shared__ float red[256];
  if (t == 0) {
    const int j = sym_idx[n];
    sh_j = j;
    sh_ck = con_arg[j];
  }
  __syncthreads();
  const int j = sh_j, ck = sh_ck;

  const float re = scb[(size_t)j * 512 + t];
  const float im = scb[(size_t)j * 512 + 256 + t];
  v2f c; c.x = re; c.y = im;
  *(v2f*)(out + O_CPLX + (size_t)n * 512 + 2 * t) = c;

  const float dr = re - zr[(size_t)n * 256 + t];
  const float di = im - zi[(size_t)n * 256 + t];
  red[t] = dr * dr + di * di;

  v4f p = {0.f, 0.f, 0.f, 0.f};
  const int base4 = 4 * t;
  if (j >= base4 && j < base4 + 4) p[j - base4] = 1.0f;
  *(v4f*)(out + O_SYMP + (size_t)n * 1024 + base4) = p;

  out[O_CONP + (size_t)n * 256 + t] = (t == ck) ? 1.0f : 0.0f;

  __syncthreads();
  for (int s = 128; s > 0; s >>= 1) { if (t < s) red[t] += red[t + s]; __syncthreads(); }
  if (t == 0) {
    part_sym[n] = red[0];
    part_con[n] = con_min[j];
    out[O_SIDX + n] = (float)j;
    out[O_CIDX + n] = (float)ck;
    out[O_CONF + n] = 1.0f / (1.0f + min_dist[n]);
  }
}

__global__ __launch_bounds__(256) void loss_reduce_k(const float* __restrict__ part_sym,
                                                     const float* __restrict__ part_con,
                                                     float* __restrict__ out) {
  __shared__ float rs[256], rc[256];
  const int t = threadIdx.x;
  float a = 0.f, b = 0.f;
  for (int i = t; i < N_ROWS; i += 256) { a += part_sym[i]; b += part_con[i]; }
  rs[t] = a; rc[t] = b;
  __syncthreads();
  for (int s = 128; s > 0; s >>= 1) {
    if (t < s) { rs[t] += rs[t + s]; rc[t] += rc[t + s]; }
    __syncthreads();
  }
  if (t == 0) {
    const float scale = 1.25f / 16777216.0f;   // 1.25 / (N_ROWS*DIM)
    out[O_LSYM] = rs[0] * scale;
    out[O_LCON] = rc[0] * scale;
  }
}

extern "C" void kernel_launch(void* const* d_in, const int* in_sizes, int n_in,
                              void* d_out, int out_size, void* d_ws, size_t ws_size,
                              hipStream_t stream) {
  const float* zr  = (const float*)d_in[0];
  const float* zi  = (const float*)d_in[1];
  const float* scb = (const float*)d_in[2];
  const float* ccb = (const float*)d_in[3];
  float* out = (float*)d_out;

  // workspace layout
  char* ws = (char*)d_ws;
  int*   w_sym_idx  = (int*)  (ws);                       // 32768
  float* w_min_dist = (float*)(ws + 131072);              // 32768
  float* w_zn2      = (float*)(ws + 262144);              // 32768
  float* w_cn2      = (float*)(ws + 393216);              // 1024
  int*   w_con_arg  = (int*)  (ws + 397312);              // 1024
  float* w_con_min  = (float*)(ws + 401408);              // 1024
  float* w_part_sym = (float*)(ws + 405504);              // 32768
  float* w_part_con = (float*)(ws + 536576);              // 32768

  row_norms_k<<<N_ROWS, 256, 0, stream>>>(zr, zi, w_zn2);
  cb_norms_k<<<K_SYM, 256, 0, stream>>>(scb, w_cn2);
  sym_gemm_k<<<N_ROWS / 32, 256, 0, stream>>>(zr, zi, scb, w_cn2, w_zn2,
                                              w_sym_idx, w_min_dist);
  concept_table_k<<<K_SYM / 16, 256, 0, stream>>>(scb, ccb, w_con_arg, w_con_min);
  scatter_out_k<<<N_ROWS, 256, 0, stream>>>(zr, zi, scb, w_sym_idx, w_min_dist,
                                            w_con_arg, w_con_min, out,
                                            w_part_sym, w_part_con);
  loss_reduce_k<<<1, 256, 0, stream>>>(w_part_sym, w_part_con, out);
}